// CrossEntropyMap_8074538517072
// MI455X (gfx1250) — compile-verified
//
#include <hip/hip_runtime.h>

typedef float f32x4 __attribute__((ext_vector_type(4)));
typedef float v2f   __attribute__((ext_vector_type(2)));
typedef float v8f   __attribute__((ext_vector_type(8)));

#define LOG2E 1.4426950408889634f
#define LN2   0.6931471805599453f

constexpr int BATCH   = 64;
constexpr int TSTEPS  = 64;
constexpr int GRID    = 128;
constexpr int ROW     = TSTEPS * GRID * GRID;   // 1,048,576 elements per batch row
constexpr int PARTS   = 16;                     // workgroups per row
constexpr int CHUNK   = ROW / PARTS;            // 65,536 elements per workgroup
constexpr int THREADS = 256;                    // 8 wave32 per workgroup
constexpr int F4_ITER = CHUNK / 4 / THREADS;    // 64 float4 loads per thread

// ---------------------------------------------------------------------------
// Kernel 1: streaming online logsumexp partials (base-2 domain).
// One block handles 256 KB of one row; emits (max, sum) partial per block.
// ---------------------------------------------------------------------------
__global__ __launch_bounds__(THREADS)
void lse_partial_kernel(const float* __restrict__ yhat,
                        float* __restrict__ partM,
                        float* __restrict__ partS) {
  const int row  = blockIdx.x / PARTS;
  const int part = blockIdx.x % PARTS;
  const int tid  = threadIdx.x;
  const f32x4* __restrict__ base =
      (const f32x4*)(yhat + (size_t)row * ROW + (size_t)part * CHUNK);

  float m = -__builtin_inff();   // running max of z = x * log2(e)
  float s = 0.0f;                // running sum of exp2(z - m)

#pragma unroll 4
  for (int j = 0; j < F4_ITER; ++j) {
    // single-use stream: non-temporal 128-bit load (global_load_b128, TH=NT)
    f32x4 v = __builtin_nontemporal_load(&base[tid + THREADS * j]);
    float z0 = v.x * LOG2E, z1 = v.y * LOG2E;
    float z2 = v.z * LOG2E, z3 = v.w * LOG2E;
    float m4 = fmaxf(fmaxf(z0, z1), fmaxf(z2, z3));
    if (m4 > m) {                          // rare after warm-up (~log n times)
      s *= __builtin_amdgcn_exp2f(m - m4);
      m  = m4;
    }
    s += __builtin_amdgcn_exp2f(z0 - m);
    s += __builtin_amdgcn_exp2f(z1 - m);
    s += __builtin_amdgcn_exp2f(z2 - m);
    s += __builtin_amdgcn_exp2f(z3 - m);
  }

  // ---- wave32 reduction: max via shuffles, then rescale ----
  float mw = m;
  for (int off = 16; off > 0; off >>= 1)
    mw = fmaxf(mw, __shfl_xor(mw, off, 32));
  float sp = s * __builtin_amdgcn_exp2f(m - mw);

  // ---- sum of 32 lane values via V_WMMA_F32_16X16X4_F32 ----
  // A (16x4): VGPR0 = sp (A[m][0]=sp_m, A[m][2]=sp_{m+16}), VGPR1 = 0.
  // B = all-ones (4x16)  =>  D[m][n] = sp_m + sp_{m+16} for every n.
  // Lanes 0-15 hold rows 0-7 in d[0..7]; lanes 16-31 hold rows 8-15.
  v2f a;     a.x = sp;   a.y = 0.0f;
  v2f bone;  bone.x = 1.0f; bone.y = 1.0f;
  v8f c = {};
  c = __builtin_amdgcn_wmma_f32_16x16x4_f32(
      /*neg_a=*/false, a, /*neg_b=*/false, bone,
      /*c_mod=*/(short)0, c, /*reuse_a=*/false, /*reuse_b=*/false);
  float p  = c[0] + c[1] + c[2] + c[3] + c[4] + c[5] + c[6] + c[7];
  float sw = p + __shfl_xor(p, 16, 32);    // full 32-lane sum on every lane

  // ---- cross-wave (8 waves) combine via LDS, fixed order (deterministic) ----
  __shared__ float lm[8], ls[8];
  const int wave = tid >> 5;
  if ((tid & 31) == 0) { lm[wave] = mw; ls[wave] = sw; }
  __syncthreads();
  if (tid == 0) {
    float M = lm[0], S = ls[0];
    for (int w = 1; w < 8; ++w) {
      float m2 = lm[w], s2 = ls[w];
      float nM = fmaxf(M, m2);
      S = S  * __builtin_amdgcn_exp2f(M  - nM)
        + s2 * __builtin_amdgcn_exp2f(m2 - nM);
      M = nM;
    }
    partM[blockIdx.x] = M;
    partS[blockIdx.x] = S;
  }
}

// ---------------------------------------------------------------------------
// Kernel 2 (single block): per-row LSE from 16 partials + picked-logit gather
// + deterministic tree reduction to the scalar loss.
// ---------------------------------------------------------------------------
__global__ __launch_bounds__(THREADS)
void finalize_kernel(const float* __restrict__ yhat,
                     const float* __restrict__ coords,
                     const float* __restrict__ partM,
                     const float* __restrict__ partS,
                     float* __restrict__ out) {
  const int tid = threadIdx.x;

  // Phase A: thread b (<64) folds the 16 partials of row b -> log2-sum-exp.
  float lse2 = 0.0f;
  if (tid < BATCH) {
    float M = -__builtin_inff(), S = 0.0f;
    for (int j = 0; j < PARTS; ++j) {
      float m2 = partM[tid * PARTS + j];
      float s2 = partS[tid * PARTS + j];
      float nM = fmaxf(M, m2);
      S = S  * __builtin_amdgcn_exp2f(M  - nM)
        + s2 * __builtin_amdgcn_exp2f(m2 - nM);
      M = nM;
    }
    lse2 = M + __builtin_amdgcn_logf(S);   // v_log_f32 (base-2)
  }

  // Phase B: gather picked logits; 4096 (b,t) pairs over 256 threads.
  float pick = 0.0f;
  for (int k = 0; k < 16; ++k) {
    int pair = tid + THREADS * k;
    int b = pair >> 6;
    int t = pair & 63;
    float cx = coords[((size_t)(b * TSTEPS + t)) * 2 + 0];
    float cy = coords[((size_t)(b * TSTEPS + t)) * 2 + 1];
    int xi = (int)rintf(cx * (float)GRID);   // v_rndne: round-half-even
    int yi = (int)rintf(cy * (float)GRID);
    size_t idx = (((size_t)(b * TSTEPS + t)) * GRID + (size_t)xi) * GRID + (size_t)yi;
    pick += yhat[idx];
  }

  // loss = -(1/B) * ( sum_pick - T * ln2 * sum_b lse2_b )
  float v = pick - (float)TSTEPS * LN2 * lse2;
  for (int off = 16; off > 0; off >>= 1)
    v += __shfl_xor(v, off, 32);
  __shared__ float red[8];
  if ((tid & 31) == 0) red[tid >> 5] = v;
  __syncthreads();
  if (tid == 0) {
    float tot = 0.0f;
    for (int w = 0; w < 8; ++w) tot += red[w];
    out[0] = -tot / (float)BATCH;
  }
}

// ---------------------------------------------------------------------------
extern "C" void kernel_launch(void* const* d_in, const int* in_sizes, int n_in,
                              void* d_out, int out_size, void* d_ws, size_t ws_size,
                              hipStream_t stream) {
  const float* yhat   = (const float*)d_in[0];  // (64,64,128,128) f32
  const float* coords = (const float*)d_in[1];  // (64,64,2) f32
  float* partM = (float*)d_ws;                  // 1024 floats
  float* partS = partM + BATCH * PARTS;         // 1024 floats (8 KB total)
  float* out   = (float*)d_out;

  lse_partial_kernel<<<BATCH * PARTS, THREADS, 0, stream>>>(yhat, partM, partS);
  finalize_kernel<<<1, THREADS, 0, stream>>>(yhat, coords, partM, partS, out);
}